// TextVideoAttentionFusion_712964571622
// MI455X (gfx1250) — compile-verified
//
#include <hip/hip_runtime.h>

// ---------------------------------------------------------------------------
// TextVideoAttentionFusion for MI455X (gfx1250, wave32)
//
// softmax over (v,h) of (score_t + score_v) == softmax(score_v): score_t is
// constant along the softmax axis, so the output is s-invariant. Compute one
// 16-row V_WMMA_F32_16X16X4_F32 tile per (b,v,f-tile) and broadcast to s=0..31.
//
// Roofline: ~103MB video read (score pass, HBM) + ~103MB (fusion pass, L2)
// + 16MB out ~= 5-6us @ 23.3TB/s. Fusion pass stages contiguous 28-row x 512-col
// chunks of the video panel into LDS (double buffered, 115KB of 320KB/WGP),
// using async global->LDS copies (ASYNCcnt) when available, so chunk c+1 copy
// overlaps the 28 WMMAs on chunk c and WMMA operands come from ds_load.
// ---------------------------------------------------------------------------

typedef float v2f __attribute__((ext_vector_type(2)));
typedef float v8f __attribute__((ext_vector_type(8)));
typedef int   v4i __attribute__((ext_vector_type(4)));
typedef __attribute__((address_space(1))) v4i g_v4i;   // global
typedef __attribute__((address_space(3))) v4i l_v4i;   // LDS

#define TF 512
#define VF 512
#define NB 4
#define NS 32
#define NV 64
#define NH 196
#define NROWS (NB * NV * NH)   // 50176 score rows
#define NPB   (NV * NH)        // 12544 softmax elements per batch

#define CK_ROWS   28                       // rows per LDS chunk (7 K-steps of 4)
#define NCHUNK    (NH / CK_ROWS)           // 7 chunks
#define CHUNK_F4  (CK_ROWS * VF / 4)       // 3584 float4 per chunk
#define F4_PER_T  (CHUNK_F4 / 256)         // 14 float4 copies per thread
#define SMEM_FLOATS (2 * CK_ROWS * VF + NH)
#define SMEM_BYTES  (SMEM_FLOATS * 4)      // 115,472 B dynamic LDS

#if __has_builtin(__builtin_amdgcn_global_load_async_to_lds_b128)
#define HAS_ASYNC_LDS 1
#else
#define HAS_ASYNC_LDS 0
#endif

// ---- Kernel 1: score_v[b,v,h] = dot(video[b,v,h,:], w[512:1024]) ----------
__global__ void k_score(const float* __restrict__ video,
                        const float* __restrict__ w,
                        float* __restrict__ score) {
    int gid  = blockIdx.x * blockDim.x + threadIdx.x;
    int row  = gid >> 5;
    int lane = threadIdx.x & 31;
    if (row >= NROWS) return;
    const float* vrow = video + (size_t)row * VF;
    const float* wv   = w + TF;
    float acc = 0.f;
#pragma unroll
    for (int j = 0; j < 4; ++j) {
        int col = j * 128 + lane * 4;
        float4 x  = *(const float4*)(vrow + col);
        float4 ww = *(const float4*)(wv + col);
        acc += x.x * ww.x + x.y * ww.y + x.z * ww.z + x.w * ww.w;
    }
#pragma unroll
    for (int m = 16; m >= 1; m >>= 1) acc += __shfl_xor(acc, m, 32);
    if (lane == 0) score[row] = acc;
}

// ---- Kernel 2: per-batch softmax over 12544 values, in place --------------
__global__ void k_softmax(float* __restrict__ s) {
    __shared__ float red[256];
    int b = blockIdx.x;
    int tid = threadIdx.x;
    float* sb = s + (size_t)b * NPB;

    float m = -3.402823466e38f;
    for (int i = tid; i < NPB; i += 256) m = fmaxf(m, sb[i]);
    red[tid] = m; __syncthreads();
    for (int off = 128; off >= 1; off >>= 1) {
        if (tid < off) red[tid] = fmaxf(red[tid], red[tid + off]);
        __syncthreads();
    }
    m = red[0]; __syncthreads();

    float sum = 0.f;
    for (int i = tid; i < NPB; i += 256) sum += __expf(sb[i] - m);
    red[tid] = sum; __syncthreads();
    for (int off = 128; off >= 1; off >>= 1) {
        if (tid < off) red[tid] += red[tid + off];
        __syncthreads();
    }
    float inv = 1.f / red[0];

    for (int i = tid; i < NPB; i += 256) sb[i] = __expf(sb[i] - m) * inv;
}

// ---- chunk copy: global (contiguous 57KB) -> LDS --------------------------
__device__ __forceinline__ void stage_chunk_async(const float* __restrict__ src,
                                                  float* __restrict__ dst,
                                                  int tid) {
#if HAS_ASYNC_LDS
#pragma unroll
    for (int i = 0; i < F4_PER_T; ++i) {
        int idx = (tid + i * 256) * 4;    // float index, 16B aligned
        __builtin_amdgcn_global_load_async_to_lds_b128(
            (g_v4i*)(src + idx),
            (l_v4i*)(dst + idx),
            0, 0);
    }
#endif
}

__device__ __forceinline__ void wait_async_lds() {
#if HAS_ASYNC_LDS
#if __has_builtin(__builtin_amdgcn_s_wait_asynccnt)
    __builtin_amdgcn_s_wait_asynccnt(0);
#else
    asm volatile("s_wait_asynccnt 0x0" ::: "memory");
#endif
#endif
}

// ---- Kernel 3: fused[b,s,v,f] = sum_h p[b,v,h] * video[b,v,h,f] -----------
// One block (256 thr, 8 waves) per (b,v). Per wave: 4 f-tiles of 16 columns.
// C(16x16) += A(16x4) * B(4x16) f32 WMMA; A rows are the (s-invariant)
// softmax weights; C rows -> s=0..15, stored twice for s=16..31.
// A layout (ISA 7.12.2): VGPR0 = K0 (lanes 0-15) / K2 (lanes 16-31), VGPR1 = K1/K3.
__global__ void k_fused(const float* __restrict__ video,
                        const float* __restrict__ p,
                        float* __restrict__ out) {
    extern __shared__ float smem[];
    float* buf0 = smem;                    // 28*512 floats
    float* buf1 = smem + CK_ROWS * VF;     // 28*512 floats
    float* lp   = smem + 2 * CK_ROWS * VF; // 196 floats

    const int bv = blockIdx.x;             // 0..255
    const int b  = bv >> 6;
    const int v  = bv & 63;
    const int tid  = threadIdx.x;
    const int wave = tid >> 5;
    const int lane = tid & 31;
    const int koff = (lane >> 4) << 1;     // 0 (lanes 0-15) or 2 (lanes 16-31)
    const int ncol = lane & 15;
    const float* V = video + (size_t)bv * NH * VF;

    if (tid < NH) lp[tid] = p[(size_t)bv * NH + tid];

    v8f acc[4];
#pragma unroll
    for (int t = 0; t < 4; ++t) acc[t] = (v8f){0.f,0.f,0.f,0.f,0.f,0.f,0.f,0.f};

#if HAS_ASYNC_LDS
    // ---------------- async double-buffered pipeline ----------------
    stage_chunk_async(V, buf0, tid);
    wait_async_lds();
    __syncthreads();
    for (int c = 0; c < NCHUNK; ++c) {
        float* cur = (c & 1) ? buf1 : buf0;
        if (c + 1 < NCHUNK)
            stage_chunk_async(V + (size_t)(c + 1) * CK_ROWS * VF,
                              (c & 1) ? buf0 : buf1, tid);
        const float* lpc = lp + c * CK_ROWS;
#pragma unroll
        for (int k = 0; k < CK_ROWS / 4; ++k) {
            v2f a;
            a.x = lpc[k * 4 + koff];
            a.y = lpc[k * 4 + koff + 1];
#pragma unroll
            for (int t = 0; t < 4; ++t) {
                int f0 = (wave + t * 8) * 16;
                const float* bp = cur + (k * 4 + koff) * VF + f0 + ncol;
                v2f bm; bm.x = bp[0]; bm.y = bp[VF];
                acc[t] = __builtin_amdgcn_wmma_f32_16x16x4_f32(
                             false, a, false, bm, (short)0, acc[t], false, false);
            }
        }
        wait_async_lds();          // chunk c+1 copies complete
        __syncthreads();
    }
#else
    // ---------------- fallback: VGPR-staged double buffer ----------------
    float4 r[F4_PER_T];
#pragma unroll
    for (int i = 0; i < F4_PER_T; ++i)
        r[i] = ((const float4*)V)[tid + i * 256];
#pragma unroll
    for (int i = 0; i < F4_PER_T; ++i)
        ((float4*)buf0)[tid + i * 256] = r[i];
    __syncthreads();
    for (int c = 0; c < NCHUNK; ++c) {
        float* cur = (c & 1) ? buf1 : buf0;
        if (c + 1 < NCHUNK) {
            const float4* src = (const float4*)(V + (size_t)(c + 1) * CK_ROWS * VF);
#pragma unroll
            for (int i = 0; i < F4_PER_T; ++i) r[i] = src[tid + i * 256];
        }
        const float* lpc = lp + c * CK_ROWS;
#pragma unroll
        for (int k = 0; k < CK_ROWS / 4; ++k) {
            v2f a;
            a.x = lpc[k * 4 + koff];
            a.y = lpc[k * 4 + koff + 1];
#pragma unroll
            for (int t = 0; t < 4; ++t) {
                int f0 = (wave + t * 8) * 16;
                const float* bp = cur + (k * 4 + koff) * VF + f0 + ncol;
                v2f bm; bm.x = bp[0]; bm.y = bp[VF];
                acc[t] = __builtin_amdgcn_wmma_f32_16x16x4_f32(
                             false, a, false, bm, (short)0, acc[t], false, false);
            }
        }
        if (c + 1 < NCHUNK) {
            float4* dst = (float4*)((c & 1) ? buf0 : buf1);
#pragma unroll
            for (int i = 0; i < F4_PER_T; ++i) dst[tid + i * 256] = r[i];
        }
        __syncthreads();
    }
#endif

    // ---- store: C reg r, lanes 0-15 -> (M=r, N), lanes 16-31 -> (M=r+8) ----
    const int srow_base = (lane >> 4) * 8;
    const size_t sstride = (size_t)NV * VF;
#pragma unroll
    for (int t = 0; t < 4; ++t) {
        int f0 = (wave + t * 8) * 16;
        size_t outbase = (size_t)b * NS * NV * VF + (size_t)v * VF + f0 + ncol;
#pragma unroll
        for (int r8 = 0; r8 < 8; ++r8) {
            int srow = srow_base + r8;
            out[outbase + (size_t)srow * sstride]        = acc[t][r8];
            out[outbase + (size_t)(srow + 16) * sstride] = acc[t][r8];
        }
    }
}

// ---------------------------------------------------------------------------
extern "C" void kernel_launch(void* const* d_in, const int* in_sizes, int n_in,
                              void* d_out, int out_size, void* d_ws, size_t ws_size,
                              hipStream_t stream) {
    // inputs: 0=text (cancels in softmax), 1=video, 2=w, 3=b (cancels)
    const float* video = (const float*)d_in[1];
    const float* w     = (const float*)d_in[2];
    float* out   = (float*)d_out;
    float* score = (float*)d_ws;   // 50176 floats = 200,704 B scratch

    k_score<<<NROWS / 8, 256, 0, stream>>>(video, w, score);
    k_softmax<<<NB, 256, 0, stream>>>(score);
    k_fused<<<NB * NV, 256, SMEM_BYTES, stream>>>(video, score, out);
}